// ResidualQuantizer_58849641890163
// MI455X (gfx1250) — compile-verified
//
#include <hip/hip_runtime.h>

// ResidualQuantizer for MI455X (gfx1250, wave32, WMMA).
// Core: per 16-row tile, S = r @ E^T via V_WMMA_F32_16X16X4_F32 with
// 4 independent WMMA chains (2 N-tiles x 2 split-K accumulators) to keep
// the matrix pipe busy; argmin over ee - 2S; residual update in registers;
// 4 stages back-to-back.

typedef float v2f __attribute__((ext_vector_type(2)));
typedef float v8f __attribute__((ext_vector_type(8)));

#define RVQ_D 32
#define RVQ_K 256
#define RVQ_L 4
#define ESTRIDE 36              // padded codebook row (floats) -> bank spread
#define WAVES_PER_BLOCK 8
#define BLOCK_THREADS (WAVES_PER_BLOCK * 32)

// dynamic LDS layout:
//   sE  : RVQ_L * RVQ_K * ESTRIDE floats (padded codebooks)
//   sEE : RVQ_L * RVQ_K floats           (per-code squared norms)
//   sIdx: WAVES_PER_BLOCK * 16 ints      (per-wave argmin broadcast)
#define SMEM_FLOATS (RVQ_L * RVQ_K * ESTRIDE + RVQ_L * RVQ_K)
#define SMEM_BYTES  (SMEM_FLOATS * 4 + WAVES_PER_BLOCK * 16 * 4)

__global__ __launch_bounds__(BLOCK_THREADS)
void rvq_kernel(const float* __restrict__ x,
                const float* __restrict__ codebooks,
                float* __restrict__ outY,
                float* __restrict__ outIdx,
                float* __restrict__ lossAcc,
                int nTiles)
{
    extern __shared__ float smem[];
    float* sE  = smem;                                   // padded codebooks
    float* sEE = sE + RVQ_L * RVQ_K * ESTRIDE;           // code norms
    int*   sIdx = (int*)(sEE + RVQ_L * RVQ_K);           // per-wave idx bcast

    const int tid = threadIdx.x;

    // ---- stage all 4 codebooks into LDS (row padded to ESTRIDE) ----
    for (int i = tid; i < RVQ_L * RVQ_K * RVQ_D; i += BLOCK_THREADS) {
        int l = i >> 13;                 // / (256*32)
        int rem = i & 8191;
        int c = rem >> 5;                // / 32
        int d = rem & 31;
        sE[(l * RVQ_K + c) * ESTRIDE + d] = codebooks[i];
    }
    __syncthreads();
    // ---- precompute ||e||^2 per code ----
    for (int i = tid; i < RVQ_L * RVQ_K; i += BLOCK_THREADS) {
        const float* row = &sE[i * ESTRIDE];
        float s = 0.f;
        #pragma unroll
        for (int d = 0; d < RVQ_D; ++d) s += row[d] * row[d];
        sEE[i] = s;
    }
    __syncthreads();

    const int lane = tid & 31;
    const int wave = tid >> 5;
    const int hi   = lane >> 4;          // half-wave select (0/1)
    const int lo   = lane & 15;
    int* myIdx = sIdx + wave * 16;

    float ssq = 0.f;                     // sum of squared per-stage residuals

    const int gw     = blockIdx.x * WAVES_PER_BLOCK + wave;
    const int stride = gridDim.x * WAVES_PER_BLOCK;

    for (int tile = gw; tile < nTiles; tile += stride) {
        const int row0 = tile * 16;

        // A-fragment layout for V_WMMA_F32_16X16X4_F32 (16x4, MxK):
        //   lane L<16 : row M=L,    K-pair base +0
        //   lane L>=16: row M=L-16, K-pair base +2
        // Fragment s covers GEMM-K dims [4s, 4s+3]; this lane owns
        // r[lo][4s+2*hi] and r[lo][4s+2*hi+1].
        v2f a[8], x0[8];
        const float* xr = x + (size_t)(row0 + lo) * RVQ_D + 2 * hi;
        if (tile + stride < nTiles)      // global_prefetch next tile's rows
            __builtin_prefetch(x + (size_t)((tile + stride) * 16 + lo) * RVQ_D, 0, 1);
        #pragma unroll
        for (int s = 0; s < 8; ++s) {
            v2f v = *(const v2f*)(xr + 4 * s);
            x0[s] = v;
            a[s]  = v;
        }

        int cidx[RVQ_L];

        #pragma unroll
        for (int l = 0; l < RVQ_L; ++l) {
            const float* E  = sE  + l * RVQ_K * ESTRIDE;
            const float* EE = sEE + l * RVQ_K;

            float best[8];
            int   bidx[8];
            #pragma unroll
            for (int j = 0; j < 8; ++j) { best[j] = 3.4e38f; bidx[j] = 0; }

            // 2 N-tiles (32 codes) per iteration, each with split-K into two
            // accumulators -> 4 independent WMMA chains of length 4.
            for (int t = 0; t < 16; t += 2) {
                v8f accA0 = {}, accA1 = {}, accB0 = {}, accB1 = {};
                // B-fragment (4x16 KxN): lane owns E[16t+lo][4s+2*hi .. +1]
                const float* EcolA = E + (t * 16 + lo) * ESTRIDE + 2 * hi;
                const float* EcolB = EcolA + 16 * ESTRIDE;
                #pragma unroll
                for (int s = 0; s < 4; ++s) {
                    v2f bA0 = *(const v2f*)(EcolA + 4 * s);
                    v2f bA1 = *(const v2f*)(EcolA + 4 * (s + 4));
                    v2f bB0 = *(const v2f*)(EcolB + 4 * s);
                    v2f bB1 = *(const v2f*)(EcolB + 4 * (s + 4));
                    accA0 = __builtin_amdgcn_wmma_f32_16x16x4_f32(
                        false, a[s],     false, bA0, (short)0, accA0, false, false);
                    accA1 = __builtin_amdgcn_wmma_f32_16x16x4_f32(
                        false, a[s + 4], false, bA1, (short)0, accA1, false, false);
                    accB0 = __builtin_amdgcn_wmma_f32_16x16x4_f32(
                        false, a[s],     false, bB0, (short)0, accB0, false, false);
                    accB1 = __builtin_amdgcn_wmma_f32_16x16x4_f32(
                        false, a[s + 4], false, bB1, (short)0, accB1, false, false);
                }
                // score = ||e||^2 - 2*S (row-constant ||r||^2 dropped)
                float eeA = EE[t * 16 + lo];
                float eeB = EE[t * 16 + 16 + lo];
                const int ciA = t * 16 + lo;
                const int ciB = ciA + 16;
                #pragma unroll
                for (int j = 0; j < 8; ++j) {        // C: row=j+8*hi, col=lo
                    float sA = eeA - 2.0f * (accA0[j] + accA1[j]);
                    float sB = eeB - 2.0f * (accB0[j] + accB1[j]);
                    bool tA = sA < best[j];
                    best[j] = tA ? sA : best[j];
                    bidx[j] = tA ? ciA : bidx[j];
                    bool tB = sB < best[j];
                    best[j] = tB ? sB : best[j];
                    bidx[j] = tB ? ciB : bidx[j];
                }
            }

            // argmin across the 16 lanes of each half (xor stays in-half);
            // branch-free selects, tie -> smaller index (matches argmin-first)
            #pragma unroll
            for (int j = 0; j < 8; ++j) {
                #pragma unroll
                for (int off = 8; off >= 1; off >>= 1) {
                    float ov = __shfl_xor(best[j], off, 32);
                    int   oi = __shfl_xor(bidx[j], off, 32);
                    bool take = (ov < best[j]) | ((ov == best[j]) & (oi < bidx[j]));
                    best[j] = take ? ov : best[j];
                    bidx[j] = take ? oi : bidx[j];
                }
            }
            // lanes 0 / 16 publish rows 0-7 / 8-15; same-wave LDS is in-order
            if (lo == 0) {
                #pragma unroll
                for (int j = 0; j < 8; ++j) myIdx[hi * 8 + j] = bidx[j];
            }
            int c = myIdx[lo];           // idx for this lane's row
            cidx[l] = c;

            // residual update + loss accumulation (r_new = r - e[c])
            const float* er = E + c * ESTRIDE + 2 * hi;
            #pragma unroll
            for (int s = 0; s < 8; ++s) {
                v2f e = *(const v2f*)(er + 4 * s);
                a[s].x -= e.x;
                a[s].y -= e.y;
                ssq += a[s].x * a[s].x + a[s].y * a[s].y;
            }
        }

        // y = x - r_final  (equals sum of selected codes, straight-through fwd)
        float* yr = outY + (size_t)(row0 + lo) * RVQ_D + 2 * hi;
        #pragma unroll
        for (int s = 0; s < 8; ++s) {
            v2f yv;
            yv.x = x0[s].x - a[s].x;
            yv.y = x0[s].y - a[s].y;
            *(v2f*)(yr + 4 * s) = yv;
        }
        // indices (B, L): low half covers all 16 rows exactly once
        if (hi == 0) {
            float* ir = outIdx + (size_t)(row0 + lo) * RVQ_L;
            #pragma unroll
            for (int l = 0; l < RVQ_L; ++l) ir[l] = (float)cidx[l];
        }
    }

    // block-free loss reduction: wave shuffle + one atomic per wave
    #pragma unroll
    for (int off = 16; off >= 1; off >>= 1) ssq += __shfl_xor(ssq, off, 32);
    if (lane == 0) atomicAdd(lossAcc, ssq);
}

__global__ void rvq_finalize(const float* __restrict__ acc,
                             float* __restrict__ outLoss,
                             float scale)
{
    *outLoss = *acc * scale;
}

extern "C" void kernel_launch(void* const* d_in, const int* in_sizes, int n_in,
                              void* d_out, int out_size, void* d_ws, size_t ws_size,
                              hipStream_t stream) {
    const float* x  = (const float*)d_in[0];   // (B, 32) f32
    const float* cb = (const float*)d_in[1];   // (4, 256, 32) f32

    const int B = in_sizes[0] / RVQ_D;         // 524288
    const int nTiles = B / 16;                 // 16 rows per wave-tile

    float* outY    = (float*)d_out;                    // B*32
    float* outIdx  = outY + (size_t)B * RVQ_D;         // B*4 (indices as f32)
    float* outLoss = outIdx + (size_t)B * RVQ_L;       // scalar
    float* lossAcc = (float*)d_ws;

    hipMemsetAsync(lossAcc, 0, sizeof(float), stream); // capture-safe memset node

    int blocks = (nTiles + WAVES_PER_BLOCK - 1) / WAVES_PER_BLOCK;
    if (blocks > 1024) blocks = 1024;

    hipLaunchKernelGGL(rvq_kernel, dim3(blocks), dim3(BLOCK_THREADS),
                       SMEM_BYTES, stream,
                       x, cb, outY, outIdx, lossAcc, nTiles);

    // mean_loss = (1/L) * sum_l [ (1+BETA)/(B*D) * ||r_l||^2 ]
    const float scale = 1.25f / (4.0f * (float)B * (float)RVQ_D);
    rvq_finalize<<<1, 1, 0, stream>>>(lossAcc, outLoss, scale);
}